// ChamferLoss_34815004901801
// MI455X (gfx1250) — compile-verified
//
#include <hip/hip_runtime.h>

typedef __attribute__((ext_vector_type(2))) float v2f;
typedef __attribute__((ext_vector_type(8))) float v8f;

#define BATCH  64
#define NPTS   1024   // N == M == 1024
#define NTILES 64     // 1024 / 16

// One workgroup per batch, 32 waves. Wave w owns row-tiles {2w, 2w+1} and
// sweeps all 64 column tiles. V_WMMA_F32_16X16X4_F32 computes the FULL
// squared distance in one op: D = (-2*P) x Q^T + (||p||^2 + ||q||^2).
__global__ __launch_bounds__(1024)
void chamfer_batch_kernel(const float* __restrict__ p,
                          const float* __restrict__ q,
                          float* __restrict__ batch_sums)
{
    __shared__ float4       qs[NPTS];      // staged q[b] rows (float4)
    __shared__ float        qnS[NPTS];     // ||q_j||^2
    __shared__ float        pnS[NPTS];     // ||p_i||^2
    __shared__ unsigned int qminU[NPTS];   // running min_i dist (uint bits, dist>=0)
    __shared__ float        red[NPTS];     // deterministic tree reduction
    __shared__ float        wsumS[32];     // per-wave row-min sums

    const int b    = blockIdx.x;
    const int tid  = threadIdx.x;
    const int wave = tid >> 5;
    const int lane = tid & 31;
    const int h    = lane >> 4;   // half-wave: 0 or 1
    const int r16  = lane & 15;

    const float* pb = p + (size_t)b * NPTS * 4;
    const float* qb = q + (size_t)b * NPTS * 4;

    // ---- Phase 0: stage q, compute norms, init column-min table ----
    {
        const float4 qv = ((const float4*)qb)[tid];
        qs[tid]  = qv;
        qnS[tid] = qv.x*qv.x + qv.y*qv.y + qv.z*qv.z + qv.w*qv.w;
        const float4 pv = ((const float4*)pb)[tid];
        pnS[tid] = pv.x*pv.x + pv.y*pv.y + pv.z*pv.z + pv.w*pv.w;
        qminU[tid] = 0x7F800000u; // +inf
    }
    __syncthreads();

    float waveSum = 0.0f;

    for (int ii = 0; ii < 2; ++ii) {
        const int itile = 2*wave + ii;
        const int i0    = itile*16 + r16;

        // A operand (16x4 f32), pre-scaled by -2:
        // lane(h,r16) holds -2*p[i0][2h], -2*p[i0][2h+1]
        const float2 a2 = *(const float2*)(pb + i0*4 + 2*h);
        v2f A; A[0] = -2.0f * a2.x; A[1] = -2.0f * a2.y;

        // ||p||^2 for the 8 rows this lane's accumulators represent (v + 8h)
        float pnr[8];
        #pragma unroll
        for (int v = 0; v < 8; ++v) pnr[v] = pnS[itile*16 + v + 8*h];

        float rowmin[8];   // per-lane running min_j of the full distance
        #pragma unroll
        for (int v = 0; v < 8; ++v) rowmin[v] = __builtin_inff();

        for (int jt = 0; jt < NTILES; ++jt) {
            const int j0 = jt*16 + r16;
            const float* qrow = (const float*)&qs[j0];
            v2f Bv; Bv[0] = qrow[2*h]; Bv[1] = qrow[2*h + 1];
            const float qnj = qnS[j0];

            // C[v] = ||p_row||^2 + ||q_col||^2  (C lane layout == our mapping)
            v8f C;
            #pragma unroll
            for (int v = 0; v < 8; ++v) C[v] = pnr[v] + qnj;

            // D = -2 p.q + ||p||^2 + ||q||^2  == squared distance
            const v8f d = __builtin_amdgcn_wmma_f32_16x16x4_f32(
                              false, A, false, Bv, (short)0, C, false, false);

            float colv = d[0];
            #pragma unroll
            for (int v = 1; v < 8; ++v) colv = fminf(colv, d[v]);
            #pragma unroll
            for (int v = 0; v < 8; ++v) rowmin[v] = fminf(rowmin[v], d[v]);

            // both half-waves contribute their 8-row column min; ds_min merges
            colv = fmaxf(colv, 0.0f);   // guard uint-ordering vs rounding
            atomicMin(&qminU[j0], __float_as_uint(colv));
        }

        // row mins: butterfly across the 16 lanes of each half-wave
        #pragma unroll
        for (int v = 0; v < 8; ++v) {
            float rv = rowmin[v];
            rv = fminf(rv, __shfl_xor(rv, 8, 32));
            rv = fminf(rv, __shfl_xor(rv, 4, 32));
            rv = fminf(rv, __shfl_xor(rv, 2, 32));
            rv = fminf(rv, __shfl_xor(rv, 1, 32));
            rowmin[v] = rv;
        }
        float s = 0.0f;
        #pragma unroll
        for (int v = 0; v < 8; ++v) s += fmaxf(rowmin[v], 0.0f);
        if (r16 == 0) waveSum += s;   // lanes 0 and 16 each carry 8 rows
    }

    waveSum += __shfl_xor(waveSum, 16, 32);
    if (lane == 0) wsumS[wave] = waveSum;
    __syncthreads();

    // ---- Phase 2: deterministic tree-sum of 1024 col-mins + 32 wave sums ----
    float val = __uint_as_float(qminU[tid]);
    if (tid < 32) val += wsumS[tid];
    red[tid] = val;
    __syncthreads();
    for (int sft = 512; sft > 0; sft >>= 1) {
        if (tid < sft) red[tid] += red[tid + sft];
        __syncthreads();
    }
    if (tid == 0) batch_sums[b] = red[0];
}

__global__ void chamfer_finalize_kernel(const float* __restrict__ batch_sums,
                                        float* __restrict__ out)
{
    __shared__ float r[BATCH];
    const int t = threadIdx.x;
    r[t] = batch_sums[t];
    __syncthreads();
    for (int s = BATCH/2; s > 0; s >>= 1) {
        if (t < s) r[t] += r[t + s];
        __syncthreads();
    }
    if (t == 0) out[0] = r[0];
}

extern "C" void kernel_launch(void* const* d_in, const int* in_sizes, int n_in,
                              void* d_out, int out_size, void* d_ws, size_t ws_size,
                              hipStream_t stream) {
    (void)in_sizes; (void)n_in; (void)out_size; (void)ws_size;
    const float* p = (const float*)d_in[0];
    const float* q = (const float*)d_in[1];
    float* out   = (float*)d_out;
    float* bsums = (float*)d_ws;   // 64 per-batch partial sums

    chamfer_batch_kernel<<<BATCH, 1024, 0, stream>>>(p, q, bsums);
    chamfer_finalize_kernel<<<1, BATCH, 0, stream>>>(bsums, out);
}